// PiggyBackBase_16466904613136
// MI455X (gfx1250) — compile-verified
//
#include <hip/hip_runtime.h>
#include <hip/hip_bf16.h>
#include <cstdint>

// NeRF/TensoRF volume-rendering composite for MI455X (gfx1250, wave32).
// One wave per ray (K=512 -> 16 samples/lane). Sigma/dists staged to LDS with
// CDNA5 async LDS loads (ASYNCcnt), rgb/z streamed directly to VGPRs.
// All streaming traffic is non-temporal (zero reuse; don't thrash 192MB L2).

#define DIST_SCALE 25.0f
#define EPS_T 1e-10f

typedef float v4f __attribute__((ext_vector_type(4)));  // true clang vector (NT-builtin OK)

__global__ __launch_bounds__(128) void nerf_composite_kernel(
    const float* __restrict__ sigma,     // [N,K]
    const float* __restrict__ rgb,       // [N,K,3]
    const float* __restrict__ dists,     // [N,K]
    const float* __restrict__ z_vals,    // [N,K]
    const float* __restrict__ rays_far,  // [N]
    float* __restrict__ rgb_map,         // [N,3]
    float* __restrict__ depth_map,       // [N]
    float* __restrict__ alpha_out,       // [N,K]
    float* __restrict__ weights_out,     // [N,K]
    float* __restrict__ bg_out,          // [N]
    int N)
{
    constexpr int K = 512;
    constexpr int WAVES = 4;             // 128 threads / wave32
    constexpr int SPL = 16;              // samples per lane (K / 32)

    __shared__ __align__(16) float sm[WAVES][2][K];  // [wave][sigma|dists][K]

    const int lane = threadIdx.x & 31;
    const int wave = threadIdx.x >> 5;
    const int ray  = blockIdx.x * WAVES + wave;
    if (ray >= N) return;                // wave-uniform guard

    const int base = ray * K + lane * SPL;          // 16 contiguous samples/lane

    // ---- CDNA5 async LDS staging of sigma & dists (ASYNCcnt path) ----
    // Low 32 bits of a generic shared pointer == DS byte address (ISA 10.2).
    unsigned ls = (unsigned)(size_t)&sm[wave][0][lane * SPL];
    unsigned ld = (unsigned)(size_t)&sm[wave][1][lane * SPL];
    const float* gs = sigma + base;
    const float* gd = dists + base;
    // INST_OFFSET is added to BOTH the global and LDS addresses (ISA 15.18.3),
    // so one LDS base VGPR covers all four 16B beats. TH=NT: single-use stream.
    asm volatile(
        "global_load_async_to_lds_b128 %0, %2, off th:TH_LOAD_NT\n\t"
        "global_load_async_to_lds_b128 %0, %2, off offset:16 th:TH_LOAD_NT\n\t"
        "global_load_async_to_lds_b128 %0, %2, off offset:32 th:TH_LOAD_NT\n\t"
        "global_load_async_to_lds_b128 %0, %2, off offset:48 th:TH_LOAD_NT\n\t"
        "global_load_async_to_lds_b128 %1, %3, off th:TH_LOAD_NT\n\t"
        "global_load_async_to_lds_b128 %1, %3, off offset:16 th:TH_LOAD_NT\n\t"
        "global_load_async_to_lds_b128 %1, %3, off offset:32 th:TH_LOAD_NT\n\t"
        "global_load_async_to_lds_b128 %1, %3, off offset:48 th:TH_LOAD_NT"
        :: "v"(ls), "v"(ld), "v"(gs), "v"(gd)
        : "memory");

    // ---- overlap: stream z and rgb straight into VGPRs (NT b128 loads) ----
    v4f zv[SPL / 4];
    const v4f* zp = (const v4f*)(z_vals + base);
#pragma unroll
    for (int i = 0; i < SPL / 4; ++i) zv[i] = __builtin_nontemporal_load(zp + i);

    v4f cv[3 * SPL / 4];                            // 48 floats of rgb
    const v4f* cp = (const v4f*)(rgb + (size_t)base * 3);
#pragma unroll
    for (int i = 0; i < 3 * SPL / 4; ++i) cv[i] = __builtin_nontemporal_load(cp + i);

    // ---- wait for async tile, then alpha / t from LDS ----
    asm volatile("s_wait_asynccnt 0x0" ::: "memory");

    float a[SPL], t[SPL];
    float p = 1.0f;                                  // per-lane product of t
    const v4f* smS = (const v4f*)&sm[wave][0][lane * SPL];
    const v4f* smD = (const v4f*)&sm[wave][1][lane * SPL];
#pragma unroll
    for (int j = 0; j < SPL / 4; ++j) {
        v4f sg4 = smS[j];
        v4f dt4 = smD[j];
#pragma unroll
        for (int u = 0; u < 4; ++u) {
            int i = 4 * j + u;
            float e = __expf(-sg4[u] * dt4[u] * DIST_SCALE);
            a[i] = 1.0f - e;
            t[i] = e + EPS_T;                        // 1 - alpha + eps
            p *= t[i];
        }
    }

    // ---- wave32 exclusive scan of per-lane products (transmittance) ----
    float incl = p;
#pragma unroll
    for (int off = 1; off < 32; off <<= 1) {
        float up = __shfl_up(incl, off, 32);
        if (lane >= off) incl *= up;
    }
    float bg = __shfl(incl, 31, 32);                 // T[:, -1] = total product
    float Texcl = __shfl_up(incl, 1, 32);
    if (lane == 0) Texcl = 1.0f;

    // ---- weights + fused reductions ----
    float w[SPL];
    float T = Texcl;
    float acc = 0.0f, dep = 0.0f, r0 = 0.0f, g0 = 0.0f, b0 = 0.0f;
    const float* zf = (const float*)zv;
    const float* cf = (const float*)cv;
#pragma unroll
    for (int i = 0; i < SPL; ++i) {
        float wi = a[i] * T;
        T *= t[i];
        w[i] = wi;
        acc += wi;
        dep += wi * zf[i];
        r0  += wi * cf[3 * i + 0];
        g0  += wi * cf[3 * i + 1];
        b0  += wi * cf[3 * i + 2];
    }

    // ---- stream alpha & weights back (NT b128 stores; never re-read) ----
    v4f* ao = (v4f*)(alpha_out + base);
    v4f* wo = (v4f*)(weights_out + base);
#pragma unroll
    for (int j = 0; j < SPL / 4; ++j) {
        v4f av = {a[4 * j], a[4 * j + 1], a[4 * j + 2], a[4 * j + 3]};
        v4f wv = {w[4 * j], w[4 * j + 1], w[4 * j + 2], w[4 * j + 3]};
        __builtin_nontemporal_store(av, ao + j);
        __builtin_nontemporal_store(wv, wo + j);
    }

    // ---- wave reduction of the 5 scalars ----
#pragma unroll
    for (int off = 16; off > 0; off >>= 1) {
        acc += __shfl_down(acc, off, 32);
        dep += __shfl_down(dep, off, 32);
        r0  += __shfl_down(r0,  off, 32);
        g0  += __shfl_down(g0,  off, 32);
        b0  += __shfl_down(b0,  off, 32);
    }

    if (lane == 0) {
        float om = 1.0f - acc;                       // white background weight
        rgb_map[ray * 3 + 0] = fminf(fmaxf(r0 + om, 0.0f), 1.0f);
        rgb_map[ray * 3 + 1] = fminf(fmaxf(g0 + om, 0.0f), 1.0f);
        rgb_map[ray * 3 + 2] = fminf(fmaxf(b0 + om, 0.0f), 1.0f);
        depth_map[ray] = dep + om * rays_far[ray];
        bg_out[ray] = bg;
    }
}

extern "C" void kernel_launch(void* const* d_in, const int* in_sizes, int n_in,
                              void* d_out, int out_size, void* d_ws, size_t ws_size,
                              hipStream_t stream) {
    (void)n_in; (void)out_size; (void)d_ws; (void)ws_size;

    const float* sigma    = (const float*)d_in[0];   // [N,K]
    const float* rgb      = (const float*)d_in[1];   // [N,K,3]
    const float* dists    = (const float*)d_in[2];   // [N,K]
    const float* z_vals   = (const float*)d_in[3];   // [N,K]
    const float* rays_far = (const float*)d_in[4];   // [N]

    const int N = in_sizes[4];                       // 32768
    const int K = in_sizes[0] / N;                   // 512 (kernel assumes 512)
    (void)K;

    // Output layout: rgb_map [N,3] | depth_map [N] | alpha [N,K] | weights [N,K] | bg [N]
    float* out       = (float*)d_out;
    float* rgb_map   = out;
    float* depth_map = rgb_map + (size_t)N * 3;
    float* alpha     = depth_map + N;
    float* weights   = alpha + (size_t)N * 512;
    float* bg        = weights + (size_t)N * 512;

    constexpr int WAVES = 4;                         // 128 threads = 4 wave32
    dim3 block(32 * WAVES);
    dim3 grid((N + WAVES - 1) / WAVES);
    nerf_composite_kernel<<<grid, block, 0, stream>>>(
        sigma, rgb, dists, z_vals, rays_far,
        rgb_map, depth_map, alpha, weights, bg, N);
}